// FastfoodWrap_4252017623140
// MI455X (gfx1250) — compile-verified
//
#include <hip/hip_runtime.h>

typedef __attribute__((ext_vector_type(2))) float v2f;
typedef __attribute__((ext_vector_type(8))) float v8f;

#define LL_BIG   (1 << 22)   // 4,194,304
#define COLS     4096        // low 12 bits (pass B, contiguous)
#define ROWS     1024        // high 10 bits (pass A, stride COLS)
#define ROWS_OUT 576         // DD_W / COLS  (2359296 = 576*4096)
#define DD_W     2359296
#define TC       32          // columns per pass-A tile
#define TSTRIDE  33          // padded LDS stride (conflict-free)
#define BPB      4           // 4096-blocks per pass-B workgroup

// ---------------- in-LDS FWHT over contiguous buffer of length n (power of 2) ---
__device__ __forceinline__ void fwht_lds(float* buf, int n, int tid, int nt) {
    for (int h = 1; h < n; h <<= 1) {
        __syncthreads();
        for (int p = tid; p < (n >> 1); p += nt) {
            int i0 = ((p & ~(h - 1)) << 1) | (p & (h - 1));
            int i1 = i0 + h;
            float a = buf[i0], b = buf[i1];
            buf[i0] = a + b;
            buf[i1] = a - b;
        }
    }
    __syncthreads();
}

// ---------------- stage 1: w = FWHT_P(BB[0:P] * theta[0:P])  (one workgroup) ----
__global__ void small_fwht_kernel(const float* __restrict__ theta,
                                  const float* __restrict__ BB,
                                  float* __restrict__ wout, int P) {
    __shared__ float w[2048];
    int tid = threadIdx.x, nt = blockDim.x;
    for (int j = tid; j < P; j += nt) w[j] = BB[j] * theta[j];
    fwht_lds(w, P, tid, nt);
    for (int j = tid; j < P; j += nt) wout[j] = w[j];
}

// ---------------- deterministic two-stage sum(GG^2) ----------------------------
__global__ void gg_partial_kernel(const float* __restrict__ GG, int n,
                                  float* __restrict__ part) {
    __shared__ float red[256];
    int tid = threadIdx.x;
    float s = 0.f;
    for (int i = blockIdx.x * blockDim.x + tid; i < n; i += gridDim.x * blockDim.x) {
        float g = GG[i]; s += g * g;
    }
    red[tid] = s; __syncthreads();
    for (int off = 128; off > 0; off >>= 1) {
        if (tid < off) red[tid] += red[tid + off];
        __syncthreads();
    }
    if (tid == 0) part[blockIdx.x] = red[0];
}

__global__ void gg_final_kernel(const float* __restrict__ part, float* __restrict__ gsum) {
    __shared__ float red[256];
    int tid = threadIdx.x;
    red[tid] = part[tid]; __syncthreads();
    for (int off = 128; off > 0; off >>= 1) {
        if (tid < off) red[tid] += red[tid + off];
        __syncthreads();
    }
    if (tid == 0) gsum[0] = red[0];
}

// ---------------- pass A: fused gather + FWHT over high 10 bits (stride 4096) --
// tile: 1024 rows x 32 cols, padded stride 33 -> 135KB LDS (of 320KB/WGP)
__global__ __launch_bounds__(512)
void fwht_passA_kernel(const float* __restrict__ w, const int* __restrict__ Pi,
                       const float* __restrict__ GG, float* __restrict__ mbuf,
                       int pmask) {
    __shared__ float tile[ROWS * TSTRIDE];
    int tid = threadIdx.x, nt = blockDim.x;
    int col0 = blockIdx.x * TC;
    // fused: m4[g] = w[Pi[g] & pmask] * GG[g]
    for (int idx = tid; idx < ROWS * TC; idx += nt) {
        int r = idx >> 5, c = idx & (TC - 1);
        int g = r * COLS + col0 + c;
        tile[r * TSTRIDE + c] = w[Pi[g] & pmask] * GG[g];
    }
    for (int h = 1; h < ROWS; h <<= 1) {
        __syncthreads();
        for (int q = tid; q < (ROWS >> 1) * TC; q += nt) {
            int c = q & (TC - 1), p = q >> 5;
            int r0 = ((p & ~(h - 1)) << 1) | (p & (h - 1));
            int r1 = r0 + h;
            float a = tile[r0 * TSTRIDE + c], b = tile[r1 * TSTRIDE + c];
            tile[r0 * TSTRIDE + c] = a + b;
            tile[r1 * TSTRIDE + c] = a - b;
        }
    }
    __syncthreads();
    // only rows < ROWS_OUT feed the first DD outputs
    for (int idx = tid; idx < ROWS_OUT * TC; idx += nt) {
        int r = idx >> 5, c = idx & (TC - 1);
        mbuf[(size_t)r * COLS + col0 + c] = tile[r * TSTRIDE + c];
    }
}

// ---------------- pass B: contiguous FWHT_4096 + fused scale/init epilogue -----
__global__ __launch_bounds__(512)
void fwht_passB_kernel(const float* __restrict__ mbuf, const float* __restrict__ init,
                       const float* __restrict__ gsum, float* __restrict__ out, int DD) {
    __shared__ float blk[BPB * COLS];   // 64KB
    int tid = threadIdx.x, nt = blockDim.x;
    size_t base = (size_t)blockIdx.x * (BPB * COLS);
    for (int i = tid; i < BPB * COLS; i += nt) blk[i] = mbuf[base + i];
    for (int h = 1; h < COLS; h <<= 1) {
        __syncthreads();
        for (int q = tid; q < BPB * (COLS >> 1); q += nt) {
            int sb = q >> 11, p = q & (COLS / 2 - 1);
            int i0 = (sb << 12) | (((p & ~(h - 1)) << 1) | (p & (h - 1)));
            int i1 = i0 + h;
            float a = blk[i0], b = blk[i1];
            blk[i0] = a + b;
            blk[i1] = a - b;
        }
    }
    __syncthreads();
    float scale = rsqrtf((float)DD * gsum[0]);
    for (int i = tid; i < BPB * COLS; i += nt) {
        size_t g = base + i;
        out[g] = init[g] + blk[i] * scale;
    }
}

// ---------------- fully-fused small fastfood (b1: LL=4096, b2: LL=1024) --------
__global__ __launch_bounds__(512)
void fastfood_small_kernel(const float* __restrict__ theta, const float* __restrict__ BB,
                           const int* __restrict__ Pi, const float* __restrict__ GG,
                           const float* __restrict__ init, float* __restrict__ out,
                           int P, int LLv, int DD) {
    __shared__ float w[2048];
    __shared__ float m[4096];
    __shared__ float red[512];
    int tid = threadIdx.x, nt = blockDim.x;
    for (int j = tid; j < P; j += nt) w[j] = BB[j] * theta[j];
    fwht_lds(w, P, tid, nt);
    float s = 0.f;
    for (int i = tid; i < LLv; i += nt) { float g = GG[i]; s += g * g; }
    red[tid] = s; __syncthreads();
    for (int off = nt >> 1; off > 0; off >>= 1) {
        if (tid < off) red[tid] += red[tid + off];
        __syncthreads();
    }
    for (int i = tid; i < LLv; i += nt) m[i] = w[Pi[i] & (P - 1)] * GG[i];
    fwht_lds(m, LLv, tid, nt);
    float scale = rsqrtf((float)DD * red[0]);
    for (int i = tid; i < DD; i += nt) out[i] = init[i] + m[i] * scale;
}

// ---------------- fp32 WMMA GEMM:  C[m,n] = act( sum_k A[m,k]*B[n,k] + bias[n] )
// One wave computes a 16x64 output strip (1 M-tile x 4 N-tiles), K in steps of 4
// via v_wmma_f32_16x16x4_f32, with 1-stage software pipelining so next-step
// operand loads are independent of (and overlap) the current step's WMMAs.
// Operand layout (16x4 f32 A / 4x16 f32 B):
//   lane l, vgpr v  ->  idx = l%16,  K = (l/16)*2 + v   (identical for A and B
//   when both are row-major with K contiguous -> plain 8B vector loads).
template<int RELU>
__global__ __launch_bounds__(256)
void gemm_wmma_kernel(const float* __restrict__ A, const float* __restrict__ B,
                      const float* __restrict__ bias, float* __restrict__ C,
                      int M, int N, int K) {
    int lane = threadIdx.x & 31;
    int wave = threadIdx.x >> 5;
    int job = blockIdx.x * 8 + wave;       // jobs = (M/16) * (N/64)
    int mTiles = M >> 4;
    int mt = job % mTiles;                 // 8 waves of a block share one ng ->
    int ng = job / mTiles;                 // B rows shared via L0/L1
    int row = lane & 15;
    int kg = (lane >> 4) << 1;             // 0 or 2
    const float* ap  = A + (size_t)(mt * 16 + row) * K + kg;
    const float* bp0 = B + (size_t)(ng * 64 + row) * K + kg;
    const float* bp1 = bp0 + (size_t)16 * K;
    const float* bp2 = bp0 + (size_t)32 * K;
    const float* bp3 = bp0 + (size_t)48 * K;
    v8f acc0 = {}, acc1 = {}, acc2 = {}, acc3 = {};

    // prologue: stage 0 operands
    v2f a  = *(const v2f*)(ap);
    v2f b0 = *(const v2f*)(bp0);
    v2f b1 = *(const v2f*)(bp1);
    v2f b2 = *(const v2f*)(bp2);
    v2f b3 = *(const v2f*)(bp3);

    for (int k0 = 4; k0 < K; k0 += 4) {
        // prefetch next k-step (independent of the WMMAs below)
        v2f an  = *(const v2f*)(ap  + k0);
        v2f bn0 = *(const v2f*)(bp0 + k0);
        v2f bn1 = *(const v2f*)(bp1 + k0);
        v2f bn2 = *(const v2f*)(bp2 + k0);
        v2f bn3 = *(const v2f*)(bp3 + k0);
        acc0 = __builtin_amdgcn_wmma_f32_16x16x4_f32(false, a, false, b0, (short)0, acc0, false, false);
        acc1 = __builtin_amdgcn_wmma_f32_16x16x4_f32(false, a, false, b1, (short)0, acc1, false, false);
        acc2 = __builtin_amdgcn_wmma_f32_16x16x4_f32(false, a, false, b2, (short)0, acc2, false, false);
        acc3 = __builtin_amdgcn_wmma_f32_16x16x4_f32(false, a, false, b3, (short)0, acc3, false, false);
        a = an; b0 = bn0; b1 = bn1; b2 = bn2; b3 = bn3;
    }
    // epilogue k-step
    acc0 = __builtin_amdgcn_wmma_f32_16x16x4_f32(false, a, false, b0, (short)0, acc0, false, false);
    acc1 = __builtin_amdgcn_wmma_f32_16x16x4_f32(false, a, false, b1, (short)0, acc1, false, false);
    acc2 = __builtin_amdgcn_wmma_f32_16x16x4_f32(false, a, false, b2, (short)0, acc2, false, false);
    acc3 = __builtin_amdgcn_wmma_f32_16x16x4_f32(false, a, false, b3, (short)0, acc3, false, false);

    // C/D layout: lane l, vgpr v -> M = v + (l/16)*8, N = l%16
    int mrow = mt * 16 + (lane >> 4) * 8;
    v8f accs[4] = {acc0, acc1, acc2, acc3};
#pragma unroll
    for (int t = 0; t < 4; ++t) {
        int n = ng * 64 + t * 16 + row;
        float bv = bias[n];
        v8f acc = accs[t];
#pragma unroll
        for (int v = 0; v < 8; ++v) {
            float xv = acc[v] + bv;
            if (RELU) xv = fmaxf(xv, 0.f);
            C[(size_t)(mrow + v) * N + n] = xv;
        }
    }
}

extern "C" void kernel_launch(void* const* d_in, const int* in_sizes, int n_in,
                              void* d_out, int out_size, void* d_ws, size_t ws_size,
                              hipStream_t stream) {
    const float* x     = (const float*)d_in[0];
    const float* theta = (const float*)d_in[1];
    const float* iW1   = (const float*)d_in[2];
    const float* ib1   = (const float*)d_in[3];
    const float* iW2   = (const float*)d_in[4];
    const float* ib2   = (const float*)d_in[5];
    const float* BB_W1 = (const float*)d_in[6];
    const int*   Pi_W1 = (const int*)d_in[7];
    const float* GG_W1 = (const float*)d_in[8];
    const float* BB_b1 = (const float*)d_in[9];
    const int*   Pi_b1 = (const int*)d_in[10];
    const float* GG_b1 = (const float*)d_in[11];
    const float* BB_W2 = (const float*)d_in[12];
    const int*   Pi_W2 = (const int*)d_in[13];
    const float* GG_W2 = (const float*)d_in[14];
    const float* BB_b2 = (const float*)d_in[15];
    const int*   Pi_b2 = (const int*)d_in[16];
    const float* GG_b2 = (const float*)d_in[17];
    float* out = (float*)d_out;

    float* ws = (float*)d_ws;
    size_t o = 0;
    float* W1   = ws + o; o += DD_W;
    float* W2   = ws + o; o += DD_W;
    float* b1   = ws + o; o += 3072;
    float* b2   = ws + o; o += 768;
    float* hbuf = ws + o; o += (size_t)256 * 3072;
    float* mbuf = ws + o; o += DD_W;
    float* wsm  = ws + o; o += 2048;
    float* part = ws + o; o += 256;
    float* gsum = ws + o; o += 1;

    // ---- W1 fastfood ----
    small_fwht_kernel<<<1, 512, 0, stream>>>(theta, BB_W1, wsm, 2048);
    gg_partial_kernel<<<256, 256, 0, stream>>>(GG_W1, LL_BIG, part);
    gg_final_kernel<<<1, 256, 0, stream>>>(part, gsum);
    fwht_passA_kernel<<<COLS / TC, 512, 0, stream>>>(wsm, Pi_W1, GG_W1, mbuf, 2047);
    fwht_passB_kernel<<<ROWS_OUT / BPB, 512, 0, stream>>>(mbuf, iW1, gsum, W1, DD_W);

    // ---- W2 fastfood (reuses wsm/part/gsum/mbuf; stream-ordered) ----
    small_fwht_kernel<<<1, 512, 0, stream>>>(theta, BB_W2, wsm, 2048);
    gg_partial_kernel<<<256, 256, 0, stream>>>(GG_W2, LL_BIG, part);
    gg_final_kernel<<<1, 256, 0, stream>>>(part, gsum);
    fwht_passA_kernel<<<COLS / TC, 512, 0, stream>>>(wsm, Pi_W2, GG_W2, mbuf, 2047);
    fwht_passB_kernel<<<ROWS_OUT / BPB, 512, 0, stream>>>(mbuf, iW2, gsum, W2, DD_W);

    // ---- b1, b2 (fully fused single-workgroup) ----
    fastfood_small_kernel<<<1, 512, 0, stream>>>(theta, BB_b1, Pi_b1, GG_b1, ib1, b1,
                                                 2048, 4096, 3072);
    fastfood_small_kernel<<<1, 512, 0, stream>>>(theta, BB_b2, Pi_b2, GG_b2, ib2, b2,
                                                 1024, 1024, 768);

    // ---- MLP: h = relu(x@W1^T + b1); out = h@W2^T + b2 ----
    // jobs = (256/16)*(3072/64) = 768 -> 96 blocks of 8 waves
    gemm_wmma_kernel<1><<<96, 256, 0, stream>>>(x, W1, b1, hbuf, 256, 3072, 768);
    // jobs = (256/16)*(768/64) = 192 -> 24 blocks
    gemm_wmma_kernel<0><<<24, 256, 0, stream>>>(hbuf, W2, b2, out, 256, 768, 3072);
}